// SE_Block_1_53953379173147
// MI455X (gfx1250) — compile-verified
//
#include <hip/hip_runtime.h>
#include <hip/hip_bf16.h>
#include <math.h>

typedef __attribute__((ext_vector_type(2))) float v2f;
typedef __attribute__((ext_vector_type(8))) float v8f;

#define C_CH   512
#define HDIM   384
#define HW     147456      // 384*384
#define HW4    36864       // HW/4
#define K1     1024        // 2*C
#define N1     256         // 2*C/r
#define N2     512         // C

// ---------------------------------------------------------------------------
// Kernel 1: per-channel single-pass online softmax moments -> feat[1024]
// one block per channel, 256 threads, each thread streams float4s.
// a = exp((x - m)/beta) with m = running max (online rescaling).
// identity = sum(a*idx)/sum(a); feat = [rho(512) | theta(512)]
// ---------------------------------------------------------------------------
__global__ __launch_bounds__(256)
void se_soft_argmax_kernel(const float* __restrict__ x,
                           const int* __restrict__ beta_p,
                           float* __restrict__ feat) {
    const int c = blockIdx.x;
    const int t = threadIdx.x;
    const float invb = 1.0f / (float)beta_p[0];

    const float4* __restrict__ xc = (const float4*)(x + (size_t)c * HW);

    float m  = -INFINITY;
    float s0 = 0.0f;
    float s1 = 0.0f;

    for (int j = t; j < HW4; j += 256) {
        const float4 v = xc[j];
        const float i0 = (float)(j * 4);
        #pragma unroll
        for (int e = 0; e < 4; ++e) {
            const float v_e = (e == 0) ? v.x : (e == 1) ? v.y : (e == 2) ? v.z : v.w;
            const float idxf = i0 + (float)e;
            if (v_e <= m) {
                const float ex = __expf((v_e - m) * invb);
                s0 += ex;
                s1 += ex * idxf;
            } else {
                const float r = __expf((m - v_e) * invb);
                s0 = s0 * r + 1.0f;
                s1 = s1 * r + idxf;
                m = v_e;
            }
        }
    }

    __shared__ float sm[256];
    __shared__ float ss0[256];
    __shared__ float ss1[256];
    sm[t] = m; ss0[t] = s0; ss1[t] = s1;
    __syncthreads();

    for (int off = 128; off > 0; off >>= 1) {
        if (t < off) {
            const float m1 = sm[t], m2 = sm[t + off];
            const float M  = fmaxf(m1, m2);
            const float r1 = __expf((m1 - M) * invb);
            const float r2 = __expf((m2 - M) * invb);
            ss0[t] = ss0[t] * r1 + ss0[t + off] * r2;
            ss1[t] = ss1[t] * r1 + ss1[t + off] * r2;
            sm[t]  = M;
        }
        __syncthreads();
    }

    if (t == 0) {
        const float identity = ss1[0] / ss0[0];
        const float col = fmodf(identity, (float)HDIM);     // identity >= 0
        const float row = truncf(identity / (float)HDIM);
        const float rho   = sqrtf(col * col + row * row);
        const float theta = atan2f(row, col) + 1e-10f;
        feat[c]         = rho;
        feat[C_CH + c]  = theta;
    }
}

// ---------------------------------------------------------------------------
// Kernel 2: SE MLP via V_WMMA_F32_16X16X4_F32 (full f32 precision).
// One workgroup, 1024 threads = 32 waves.
//   layer1: h[256]  = relu(feat[1024] @ w1[1024,256])  -> waves 0..15, K=1024
//   layer2: y[512]  = sigmoid(h[256]  @ w2[256, 512])  -> waves 0..31, K=256
// A fragment broadcasts the feature row across all 16 M rows; row 0 is kept.
// ISA striping: A/B VGPR0 holds K rows {0,2} across lane halves, VGPR1 {1,3}.
// ---------------------------------------------------------------------------
__global__ __launch_bounds__(1024)
void se_mlp_wmma_kernel(const float* __restrict__ feat,
                        const float* __restrict__ w1,
                        const float* __restrict__ w2,
                        float* __restrict__ y) {
    __shared__ float feat_lds[K1];
    __shared__ float h_lds[N1];

    const int t    = threadIdx.x;
    const int wave = t >> 5;
    const int lane = t & 31;
    const int n    = lane & 15;          // N index within 16-wide tile
    const int kh   = (lane >> 4) << 1;   // 0 for lanes 0-15, 2 for lanes 16-31

    if (t < K1) feat_lds[t] = feat[t];
    __syncthreads();

    // ---- layer 1: 16 N-tiles, K = 1024 (256 wmma steps) ----
    if (wave < 16) {
        const int ncol = wave * 16 + n;
        v8f acc = {};
        for (int k = 0; k < K1; k += 4) {
            v2f a, b;
            a.x = feat_lds[k + kh + 0];
            a.y = feat_lds[k + kh + 1];
            b.x = w1[(size_t)(k + kh + 0) * N1 + ncol];
            b.y = w1[(size_t)(k + kh + 1) * N1 + ncol];
            acc = __builtin_amdgcn_wmma_f32_16x16x4_f32(
                      false, a, false, b, (short)0, acc, false, false);
        }
        if (lane < 16) h_lds[ncol] = fmaxf(acc[0], 0.0f);   // row M=0, relu
    }
    __syncthreads();

    // ---- layer 2: 32 N-tiles, K = 256 (64 wmma steps) ----
    {
        const int ncol = wave * 16 + n;
        v8f acc = {};
        for (int k = 0; k < N1; k += 4) {
            v2f a, b;
            a.x = h_lds[k + kh + 0];
            a.y = h_lds[k + kh + 1];
            b.x = w2[(size_t)(k + kh + 0) * N2 + ncol];
            b.y = w2[(size_t)(k + kh + 1) * N2 + ncol];
            acc = __builtin_amdgcn_wmma_f32_16x16x4_f32(
                      false, a, false, b, (short)0, acc, false, false);
        }
        if (lane < 16) y[ncol] = 1.0f / (1.0f + __expf(-acc[0]));  // sigmoid
    }
}

// ---------------------------------------------------------------------------
// Kernel 3: out = x * y[c], pure streaming float4 pass.
// 1024 elements per block -> 144 blocks per channel (exact), so the channel
// id (and thus y[c]) is block-uniform -> scalar load.
// ---------------------------------------------------------------------------
__global__ __launch_bounds__(256)
void se_scale_kernel(const float* __restrict__ x,
                     const float* __restrict__ y,
                     float* __restrict__ out) {
    const float s = y[blockIdx.x / 144];          // uniform per block
    const size_t j = (size_t)blockIdx.x * 256 + threadIdx.x;   // float4 index
    const float4 v = ((const float4*)x)[j];
    float4 r;
    r.x = v.x * s; r.y = v.y * s; r.z = v.z * s; r.w = v.w * s;
    ((float4*)out)[j] = r;
}

extern "C" void kernel_launch(void* const* d_in, const int* in_sizes, int n_in,
                              void* d_out, int out_size, void* d_ws, size_t ws_size,
                              hipStream_t stream) {
    const float* x    = (const float*)d_in[0];   // [1,512,384,384]
    const float* w1   = (const float*)d_in[1];   // [1024,256] (in,out)
    const float* w2   = (const float*)d_in[2];   // [256,512]  (in,out)
    const int*   beta = (const int*)d_in[3];     // scalar (2)
    float* out = (float*)d_out;

    float* feat = (float*)d_ws;        // 1024 floats
    float* y    = feat + K1;           // 512 floats

    se_soft_argmax_kernel<<<dim3(C_CH), dim3(256), 0, stream>>>(x, beta, feat);
    se_mlp_wmma_kernel<<<dim3(1), dim3(1024), 0, stream>>>(feat, w1, w2, y);
    se_scale_kernel<<<dim3(C_CH * 144), dim3(256), 0, stream>>>(x, y, out);
}